// GAT_54116587929922
// MI455X (gfx1250) — compile-verified
//
#include <hip/hip_runtime.h>

typedef __attribute__((ext_vector_type(2))) float v2f;
typedef __attribute__((ext_vector_type(8))) float v8f;

#define N_NODES 50000
#define N_EDGES 800000
#define DFEAT   128     // H*D_HEAD == D_IN == 128
#define NHEAD   4
#define DHEAD   32
#define NEG_SLOPE 0.2f

// ---------------- monotone float<->uint key for atomic max ----------------
__device__ __forceinline__ unsigned fkey(float f) {
  unsigned b = __float_as_uint(f);
  return (b & 0x80000000u) ? ~b : (b | 0x80000000u);
}
__device__ __forceinline__ float fdecode(unsigned k) {
  unsigned b = (k & 0x80000000u) ? (k & 0x7FFFFFFFu) : ~k;
  return __uint_as_float(b);
}

// ---------------- edge-weighted aggregation: acc[dst] += h[src]*w ----------
// one wave (32 lanes) per edge; each lane moves a float4 (32*4 = 128 feats)
__global__ void k_edge_weighted_agg(const float* __restrict__ h,
                                    const float* __restrict__ w,
                                    const int* __restrict__ src,
                                    const int* __restrict__ dst,
                                    float* __restrict__ acc, int nE) {
  int e = blockIdx.x * (blockDim.x >> 5) + (threadIdx.x >> 5);
  if (e >= nE) return;
  int lane = threadIdx.x & 31;
  int s = src[e], d = dst[e];
  float wt = w[e];
  float4 v = ((const float4*)(h + (size_t)s * DFEAT))[lane];
  float* o = acc + (size_t)d * DFEAT + lane * 4;
  atomicAdd(o + 0, v.x * wt);
  atomicAdd(o + 1, v.y * wt);
  atomicAdd(o + 2, v.z * wt);
  atomicAdd(o + 3, v.w * wt);
}

// ---------------- dense GEMM C = A(N x128) * B(128x128) via f32 WMMA -------
// block = 256 threads = 8 waves; block owns 16 rows; wave w owns cols 16w..16w+15
// A tile (16x128) staged in LDS; K stepped by 4 with V_WMMA_F32_16X16X4_F32
__global__ void k_gemm128_wmma(const float* __restrict__ A,
                               const float* __restrict__ B,
                               float* __restrict__ C) {
  __shared__ float As[16 * DFEAT];           // 8 KB
  const int row0 = blockIdx.x * 16;
  const int tid  = threadIdx.x;

  // cooperative load of 16x128 A tile (2048 floats, float4 vectorized)
  for (int i = tid; i < 16 * DFEAT / 4; i += 256) {
    ((float4*)As)[i] = ((const float4*)(A + (size_t)row0 * DFEAT))[i];
  }
  __syncthreads();

  const int wave = tid >> 5;                 // 0..7 -> output col block
  const int lane = tid & 31;
  const int col0 = wave * 16;
  const int mn   = lane & 15;                // M index (A) / N index (B)
  const int kh   = (lane >> 4) * 2;          // lanes 16-31 hold K+2/K+3

  v8f acc = {};
  for (int k = 0; k < DFEAT; k += 4) {
    v2f a, b;
    a.x = As[mn * DFEAT + k + kh];
    a.y = As[mn * DFEAT + k + kh + 1];
    b.x = B[(size_t)(k + kh)     * DFEAT + col0 + mn];
    b.y = B[(size_t)(k + kh + 1) * DFEAT + col0 + mn];
    acc = __builtin_amdgcn_wmma_f32_16x16x4_f32(false, a, false, b,
                                                (short)0, acc, false, false);
  }

  // C/D layout: VGPR r -> (M = r + 8*(lane>=16), N = lane&15)
  const int rbase = (lane >> 4) * 8;
  const int col   = col0 + mn;
  for (int r = 0; r < 8; ++r) {
    C[(size_t)(row0 + rbase + r) * DFEAT + col] = acc[r];
  }
}

// ---------------- per-(node,head) attention dots --------------------------
__global__ void k_attn_scores(const float* __restrict__ feat,
                              const float* __restrict__ al,
                              const float* __restrict__ ar,
                              float* __restrict__ el, float* __restrict__ er,
                              int n) {
  int idx = blockIdx.x * blockDim.x + threadIdx.x;   // node*4 + head
  if (idx >= n * NHEAD) return;
  int node = idx >> 2, hh = idx & 3;
  const float* f  = feat + (size_t)node * DFEAT + hh * DHEAD;
  const float* a1 = al + hh * DHEAD;
  const float* a2 = ar + hh * DHEAD;
  float sl = 0.f, sr = 0.f;
  #pragma unroll
  for (int d = 0; d < DHEAD; ++d) { float v = f[d]; sl += v * a1[d]; sr += v * a2[d]; }
  el[idx] = sl; er[idx] = sr;
}

// ---------------- edge softmax: pass 1, segment max -----------------------
__global__ void k_edge_max(const float* __restrict__ el, const float* __restrict__ er,
                           const int* __restrict__ src, const int* __restrict__ dst,
                           unsigned* __restrict__ mkey, int nE) {
  int idx = blockIdx.x * blockDim.x + threadIdx.x;   // edge*4 + head
  if (idx >= nE * NHEAD) return;
  int e = idx >> 2, hh = idx & 3;
  float x = el[src[e] * NHEAD + hh] + er[dst[e] * NHEAD + hh];
  x = (x > 0.f) ? x : NEG_SLOPE * x;                 // leaky_relu
  atomicMax(mkey + dst[e] * NHEAD + hh, fkey(x));
}

// ---------------- edge softmax: pass 2, exp + segment sum -----------------
__global__ void k_edge_expsum(const float* __restrict__ el, const float* __restrict__ er,
                              const int* __restrict__ src, const int* __restrict__ dst,
                              const unsigned* __restrict__ mkey,
                              float* __restrict__ ee, float* __restrict__ ssum, int nE) {
  int idx = blockIdx.x * blockDim.x + threadIdx.x;
  if (idx >= nE * NHEAD) return;
  int e = idx >> 2, hh = idx & 3;
  int d = dst[e];
  float x = el[src[e] * NHEAD + hh] + er[d * NHEAD + hh];
  x = (x > 0.f) ? x : NEG_SLOPE * x;
  float m = fdecode(mkey[d * NHEAD + hh]);
  float v = expf(x - m);
  ee[idx] = v;
  atomicAdd(ssum + d * NHEAD + hh, v);
}

// ---------------- attention-weighted message aggregation ------------------
// wave per edge: rst[dst] += feat[src] * alpha(head)
__global__ void k_edge_msg(const float* __restrict__ feat,
                           const int* __restrict__ src, const int* __restrict__ dst,
                           const float* __restrict__ ee, const float* __restrict__ ssum,
                           float* __restrict__ rst, int nE) {
  int e = blockIdx.x * (blockDim.x >> 5) + (threadIdx.x >> 5);
  if (e >= nE) return;
  int lane = threadIdx.x & 31;
  int s = src[e], d = dst[e];
  int hh = lane >> 3;                                // feature 4*lane is in head lane/8
  float alpha = ee[e * NHEAD + hh] / ssum[d * NHEAD + hh];
  float4 v = ((const float4*)(feat + (size_t)s * DFEAT))[lane];
  float* o = rst + (size_t)d * DFEAT + lane * 4;
  atomicAdd(o + 0, v.x * alpha);
  atomicAdd(o + 1, v.y * alpha);
  atomicAdd(o + 2, v.z * alpha);
  atomicAdd(o + 3, v.w * alpha);
}

// ---------------- layer-0 epilogue: h1 = relu(h1 + b + residual) ----------
__global__ void k_finalize0(float* __restrict__ h1, const float* __restrict__ b,
                            const float* __restrict__ resid, int n) {
  int i = blockIdx.x * blockDim.x + threadIdx.x;
  if (i >= n) return;
  float v = h1[i] + b[i & (DFEAT - 1)] + resid[i];
  h1[i] = (v > 0.f) ? v : 0.f;
}

// ---------------- layer-1 epilogue: out += b ------------------------------
__global__ void k_finalize1(float* __restrict__ out, const float* __restrict__ b, int n) {
  int i = blockIdx.x * blockDim.x + threadIdx.x;
  if (i >= n) return;
  out[i] += b[i & (DFEAT - 1)];
}

extern "C" void kernel_launch(void* const* d_in, const int* in_sizes, int n_in,
                              void* d_out, int out_size, void* d_ws, size_t ws_size,
                              hipStream_t stream) {
  (void)in_sizes; (void)n_in; (void)out_size; (void)ws_size;

  const float* in_feat = (const float*)d_in[0];
  const float* ew      = (const float*)d_in[1];
  const float* W0      = (const float*)d_in[2];
  const float* al0     = (const float*)d_in[3];
  const float* ar0     = (const float*)d_in[4];
  const float* b0      = (const float*)d_in[5];
  const float* W1      = (const float*)d_in[6];
  const float* al1     = (const float*)d_in[7];
  const float* ar1     = (const float*)d_in[8];
  const float* b1      = (const float*)d_in[9];
  const int*   src     = (const int*)d_in[10];
  const int*   dst     = (const int*)d_in[11];
  float*       out     = (float*)d_out;

  const size_t N = N_NODES, E = N_EDGES;
  const size_t nodeFeats = N * DFEAT;

  // workspace layout (floats)
  float*    hw   = (float*)d_ws;            // N*128   GEMM input (post weighted agg)
  float*    feat = hw   + nodeFeats;        // N*128   GEMM output
  float*    h1   = feat + nodeFeats;        // N*128   layer-0 result / rst accumulator
  float*    el   = h1   + nodeFeats;        // N*4
  float*    er   = el   + N * NHEAD;        // N*4
  unsigned* mkey = (unsigned*)(er + N * NHEAD); // N*4
  float*    ssum = (float*)mkey + N * NHEAD;    // N*4
  float*    ee   = ssum + N * NHEAD;        // E*4

  const int EW_BLOCKS = (int)((E + 7) / 8); // wave-per-edge, 8 waves/block
  const int EH = (int)(E * NHEAD);

  // ===================== layer 0 =====================
  hipMemcpyAsync(hw, in_feat, nodeFeats * sizeof(float), hipMemcpyDeviceToDevice, stream);
  k_edge_weighted_agg<<<EW_BLOCKS, 256, 0, stream>>>(in_feat, ew, src, dst, hw, (int)E);
  k_gemm128_wmma<<<(int)(N / 16), 256, 0, stream>>>(hw, W0, feat);
  k_attn_scores<<<(int)((N * NHEAD + 255) / 256), 256, 0, stream>>>(feat, al0, ar0, el, er, (int)N);
  hipMemsetAsync(mkey, 0, N * NHEAD * sizeof(unsigned), stream);
  hipMemsetAsync(ssum, 0, N * NHEAD * sizeof(float), stream);
  k_edge_max<<<(EH + 255) / 256, 256, 0, stream>>>(el, er, src, dst, mkey, (int)E);
  k_edge_expsum<<<(EH + 255) / 256, 256, 0, stream>>>(el, er, src, dst, mkey, ee, ssum, (int)E);
  hipMemsetAsync(h1, 0, nodeFeats * sizeof(float), stream);
  k_edge_msg<<<EW_BLOCKS, 256, 0, stream>>>(feat, src, dst, ee, ssum, h1, (int)E);
  k_finalize0<<<(int)((nodeFeats + 255) / 256), 256, 0, stream>>>(h1, b0, in_feat, (int)nodeFeats);

  // ===================== layer 1 =====================
  hipMemcpyAsync(hw, h1, nodeFeats * sizeof(float), hipMemcpyDeviceToDevice, stream);
  k_edge_weighted_agg<<<EW_BLOCKS, 256, 0, stream>>>(h1, ew, src, dst, hw, (int)E);
  k_gemm128_wmma<<<(int)(N / 16), 256, 0, stream>>>(hw, W1, feat);
  k_attn_scores<<<(int)((N * NHEAD + 255) / 256), 256, 0, stream>>>(feat, al1, ar1, el, er, (int)N);
  hipMemsetAsync(mkey, 0, N * NHEAD * sizeof(unsigned), stream);
  hipMemsetAsync(ssum, 0, N * NHEAD * sizeof(float), stream);
  k_edge_max<<<(EH + 255) / 256, 256, 0, stream>>>(el, er, src, dst, mkey, (int)E);
  k_edge_expsum<<<(EH + 255) / 256, 256, 0, stream>>>(el, er, src, dst, mkey, ee, ssum, (int)E);
  hipMemsetAsync(out, 0, nodeFeats * sizeof(float), stream);
  k_edge_msg<<<EW_BLOCKS, 256, 0, stream>>>(feat, src, dst, ee, ssum, out, (int)E);
  k_finalize1<<<(int)((nodeFeats + 255) / 256), 256, 0, stream>>>(out, b1, (int)nodeFeats);
}